// MaskedMaxPool_51780125721460
// MI455X (gfx1250) — compile-verified
//
#include <hip/hip_runtime.h>

// PointNet++ set abstraction: FPS -> ball query (WMMA distances) -> gather+maxpool
// Sizes fixed by the reference.
#define BATCH 16
#define NPTS  4096
#define NCH   128
#define NSAMP 1024
#define KNB   32
#define RAD2  0.04f

#define OUT_XYZ_ELEMS (BATCH * NSAMP * 3)

typedef float v2f __attribute__((ext_vector_type(2)));
typedef float v8f __attribute__((ext_vector_type(8)));

// ---------------------------------------------------------------------------
// Kernel 1: farthest point sampling. One block per batch, 1024 threads
// (32 waves), each thread owns 4 points with running min-distance in
// registers; xyz cached in LDS (48KB). Argmax per step: wave32 shfl-xor
// reduce + 32-entry LDS reduce (2 barriers/step). Writes new_xyz directly
// into d_out (row = coords of the selected centroid, recorded BEFORE the
// distance update, matching the jax scan semantics).
// ---------------------------------------------------------------------------
__global__ void __launch_bounds__(1024) fps_kernel(const float* __restrict__ xyz,
                                                   float* __restrict__ out_newxyz) {
  __shared__ float sx[NPTS], sy[NPTS], sz[NPTS];
  __shared__ float redD[32];
  __shared__ int   redI[32];
  __shared__ int   sFar;

  const int b   = blockIdx.x;
  const int tid = threadIdx.x;
  const float* xb = xyz + (size_t)b * NPTS * 3;

  for (int i = tid; i < NPTS; i += 1024) {
    sx[i] = xb[i * 3 + 0];
    sy[i] = xb[i * 3 + 1];
    sz[i] = xb[i * 3 + 2];
  }
  __syncthreads();

  float px[4], py[4], pz[4], dist[4];
#pragma unroll
  for (int j = 0; j < 4; ++j) {
    int i = tid + 1024 * j;
    px[j] = sx[i]; py[j] = sy[i]; pz[j] = sz[i];
    dist[j] = 1.0e10f;
  }

  const int lane = tid & 31;
  const int wid  = tid >> 5;
  int far = 0;
  float* nxyz = out_newxyz + (size_t)b * NSAMP * 3;

  for (int it = 0; it < NSAMP; ++it) {
    const float cx = sx[far], cy = sy[far], cz = sz[far];
    if (tid == 0) {
      nxyz[it * 3 + 0] = cx;
      nxyz[it * 3 + 1] = cy;
      nxyz[it * 3 + 2] = cz;
    }
    float bd = -1.0f; int bi = 0x7fffffff;
#pragma unroll
    for (int j = 0; j < 4; ++j) {
      float dx = px[j] - cx, dy = py[j] - cy, dz = pz[j] - cz;
      float d  = dx * dx + dy * dy + dz * dz;
      dist[j]  = fminf(dist[j], d);
      int i = tid + 1024 * j;
      if (dist[j] > bd || (dist[j] == bd && i < bi)) { bd = dist[j]; bi = i; }
    }
    // wave32 argmax reduce (lowest index wins ties, like jnp.argmax)
#pragma unroll
    for (int m = 16; m >= 1; m >>= 1) {
      float od = __shfl_xor(bd, m, 32);
      int   oi = __shfl_xor(bi, m, 32);
      if (od > bd || (od == bd && oi < bi)) { bd = od; bi = oi; }
    }
    if (lane == 0) { redD[wid] = bd; redI[wid] = bi; }
    __syncthreads();
    if (wid == 0) {
      bd = redD[lane]; bi = redI[lane];
#pragma unroll
      for (int m = 16; m >= 1; m >>= 1) {
        float od = __shfl_xor(bd, m, 32);
        int   oi = __shfl_xor(bi, m, 32);
        if (od > bd || (od == bd && oi < bi)) { bd = od; bi = oi; }
      }
      if (lane == 0) sFar = bi;
    }
    __syncthreads();
    far = sFar;
  }
}

// ---------------------------------------------------------------------------
// Kernel 2: transpose features (B,128,4096) -> featT (B,4096,128) so that
// every neighbor gather in kernel 4 is a contiguous 512B row. Classic 32x32
// LDS tile with +1 padding; both global read and write fully coalesced.
// ---------------------------------------------------------------------------
__global__ void __launch_bounds__(256) transpose_kernel(const float* __restrict__ f,
                                                        float* __restrict__ ft) {
  __shared__ float t[32][33];
  const int b  = blockIdx.z;
  const int n0 = blockIdx.x * 32;
  const int c0 = blockIdx.y * 32;
  const float* fb  = f  + (size_t)b * NCH * NPTS;
  float*       ftb = ft + (size_t)b * NPTS * NCH;

#pragma unroll
  for (int j = 0; j < 32; j += 8)
    t[threadIdx.y + j][threadIdx.x] =
        fb[(size_t)(c0 + threadIdx.y + j) * NPTS + n0 + threadIdx.x];
  __syncthreads();
#pragma unroll
  for (int j = 0; j < 32; j += 8)
    ftb[(size_t)(n0 + threadIdx.y + j) * NCH + c0 + threadIdx.x] =
        t[threadIdx.x][threadIdx.y + j];
}

// ---------------------------------------------------------------------------
// Kernel 3: ball query. The pairwise dot products of _square_distance are a
// (S x 3) @ (3 x N) GEMM -> V_WMMA_F32_16X16X4_F32 (K padded 3->4 with 0).
// One wave owns 16 sampled points. Per 16-point chunk:
//   - B operand (4x16) built from prefetched coords (lanes 0-15: x/y = K0/K1,
//     lanes 16-31: z/0 = K2/K3, column N = lane&15), same split for A.
//   - one WMMA produces the 16x16 dot tile in registers:
//     Cv[r] = D[M = r + 8*(lane>=16)][N = lane&15].
//   - d^2 = sn[row] + nn[col] - 2*dot assembled fully in registers
//     (sn via 8 pre-loop shuffles, nn from the coords already in hand),
//     then ONE wave32 ballot per r yields two complete 16-bit row masks.
//   - lane m extracts its row mask via a 3-level uniform mux and appends
//     only set bits (ctz loop) in ascending scan order == the reference's
//     "32 smallest indices within radius"; pads with the first neighbor
//     (always exists: each sample is its own neighbor at d^2 = 0).
//   - early exit once every row has K neighbors.
// No LDS, no barriers, no dscnt stalls.
// ---------------------------------------------------------------------------
#define BQ_WAVES 8
__global__ void __launch_bounds__(BQ_WAVES * 32)
ballquery_kernel(const float* __restrict__ xyz,
                 const float* __restrict__ new_xyz,
                 int* __restrict__ ball_idx) {
  const int b     = blockIdx.y;
  const int sTile = blockIdx.x;                  // NSAMP / (BQ_WAVES*16) tiles
  const int tid   = threadIdx.x;
  const int lane  = tid & 31;
  const int w     = tid >> 5;
  const int m     = lane & 15;
  const int half  = lane >> 4;

  const float* xb = xyz + (size_t)b * NPTS * 3;
  const int s = sTile * (BQ_WAVES * 16) + w * 16 + m;
  const float* sp = new_xyz + ((size_t)b * NSAMP + s) * 3;
  const float ax = sp[0], ay = sp[1], az = sp[2];
  const float snorm = ax * ax + ay * ay + az * az;

  // A matrix (16x4 f32): lanes 0-15 hold K=0 (x) / K=1 (y); lanes 16-31 hold
  // K=2 (z) / K=3 (zero pad), row M = lane%16.
  v2f A;
  A.x = half ? az : ax;
  A.y = half ? 0.0f : ay;

  // Row norms for the 8 D rows this lane owns (loop-invariant).
  float sn[8];
#pragma unroll
  for (int r = 0; r < 8; ++r) sn[r] = __shfl(snorm, r + 8 * half, 32);

  int cnt = 0, first = 0;
  int* outIdx = ball_idx + ((size_t)b * NSAMP + s) * KNB;

  // Software-pipelined coordinate fetch (column n0 + m).
  float nx = xb[m * 3 + 0], ny = xb[m * 3 + 1], nz = xb[m * 3 + 2];

  for (int n0 = 0; n0 < NPTS; n0 += 16) {
    const int np = (n0 + 16 < NPTS) ? (n0 + 16 + m) : m;
    const float fx = xb[np * 3 + 0];     // prefetch next chunk
    const float fy = xb[np * 3 + 1];
    const float fz = xb[np * 3 + 2];

    // B matrix (4x16 f32): same lane split as A, column N = lane%16.
    v2f Bv;
    Bv.x = half ? nz : nx;
    Bv.y = half ? 0.0f : ny;
    v8f Cv = {0.f, 0.f, 0.f, 0.f, 0.f, 0.f, 0.f, 0.f};
    Cv = __builtin_amdgcn_wmma_f32_16x16x4_f32(
        /*neg_a=*/false, A, /*neg_b=*/false, Bv,
        /*c_mod=*/(short)0, Cv, /*reuse_a=*/false, /*reuse_b=*/false);

    const float nn = nx * nx + ny * ny + nz * nz;   // column norm, in hand

    unsigned rm[8];                                  // wave-uniform ballots
#pragma unroll
    for (int r = 0; r < 8; ++r) {
      const float d2 = sn[r] + nn - 2.0f * Cv[r];
      rm[r] = (unsigned)__ballot(d2 <= RAD2);        // bits0-15: row r, bits16-31: row r+8
    }

    // Lane m selects its row mask: 3-level mux over uniform values.
    const unsigned a0 = (lane & 1) ? rm[1] : rm[0];
    const unsigned a1 = (lane & 1) ? rm[3] : rm[2];
    const unsigned a2 = (lane & 1) ? rm[5] : rm[4];
    const unsigned a3 = (lane & 1) ? rm[7] : rm[6];
    const unsigned b0 = (lane & 2) ? a1 : a0;
    const unsigned b1 = (lane & 2) ? a3 : a2;
    const unsigned c0 = (lane & 4) ? b1 : b0;
    unsigned mask = (lane & 8) ? (c0 >> 16) : (c0 & 0xFFFFu);

    if (lane < 16) {
      while (mask) {
        const int j = __builtin_ctz(mask);
        mask &= mask - 1;
        const int n = n0 + j;
        if (cnt == 0) first = n;
        if (cnt < KNB) outIdx[cnt] = n;
        ++cnt;
      }
    }

    // All rows saturated? Nothing left to store.
    if (__ballot((lane < 16) && (cnt < KNB)) == 0) break;

    nx = fx; ny = fy; nz = fz;
  }

  if (lane < 16) {
    int c = cnt < KNB ? cnt : KNB;
    for (int k = c; k < KNB; ++k) outIdx[k] = first;  // pad with group_first
  }
}

// ---------------------------------------------------------------------------
// Kernel 4: fused gather + max-pool. Block = 256 threads handles 32 samples.
// Neighbor index tile preloaded to LDS; each gather reads a contiguous
// 128-float row of featT (coalesced across the 128 channel lanes, mostly L2
// hits since featT fits in the 192MB L2). Results staged in LDS so the
// (B,C,S) output stores are coalesced along S.
// ---------------------------------------------------------------------------
__global__ void __launch_bounds__(256)
gather_maxpool_kernel(const float* __restrict__ featT,
                      const int* __restrict__ ball_idx,
                      float* __restrict__ out) {
  __shared__ int   idxT[32 * KNB];     // 4KB
  __shared__ float ot[NCH * 33];       // [c][s], stride 33 to dodge conflicts

  const int b   = blockIdx.y;
  const int s0  = blockIdx.x * 32;
  const int tid = threadIdx.x;

  const int* ib = ball_idx + ((size_t)b * NSAMP + s0) * KNB;
  for (int i = tid; i < 32 * KNB; i += 256) idxT[i] = ib[i];
  __syncthreads();

  const float* ftb = featT + (size_t)b * NPTS * NCH;
  const int c  = tid & 127;
  const int sl = tid >> 7;
  for (int s = sl; s < 32; s += 2) {
    const int* row = idxT + s * KNB;
    float mx = -3.402823466e38f;
#pragma unroll 4
    for (int k = 0; k < KNB; ++k) {
      mx = fmaxf(mx, ftb[(size_t)row[k] * NCH + c]);
    }
    ot[c * 33 + s] = mx;
  }
  __syncthreads();

  float* ob = out + OUT_XYZ_ELEMS + (size_t)b * NCH * NSAMP;
  for (int i = tid; i < NCH * 32; i += 256) {
    const int cc = i >> 5, ss = i & 31;
    ob[(size_t)cc * NSAMP + s0 + ss] = ot[cc * 33 + ss];
  }
}

// ---------------------------------------------------------------------------
extern "C" void kernel_launch(void* const* d_in, const int* in_sizes, int n_in,
                              void* d_out, int out_size, void* d_ws, size_t ws_size,
                              hipStream_t stream) {
  (void)in_sizes; (void)n_in; (void)out_size; (void)ws_size;
  const float* xyz  = (const float*)d_in[0];  // (16, 4096, 3)
  const float* feat = (const float*)d_in[1];  // (16, 128, 4096)
  float* out = (float*)d_out;                 // new_xyz (49152) ++ sub_features (2097152)

  // Workspace layout: [ ball_idx: 16*1024*32 int = 2MB ][ featT: 16*4096*128 f32 = 32MB ]
  int*   ball_idx = (int*)d_ws;
  float* featT    = (float*)((char*)d_ws + (size_t)BATCH * NSAMP * KNB * sizeof(int));

  // 1) FPS -> writes new_xyz into the head of d_out
  fps_kernel<<<BATCH, 1024, 0, stream>>>(xyz, out);

  // 2) features transpose (independent of FPS; stream-ordered anyway)
  transpose_kernel<<<dim3(NPTS / 32, NCH / 32, BATCH), dim3(32, 8), 0, stream>>>(feat, featT);

  // 3) ball query via WMMA distance tiles (consumes new_xyz from d_out)
  ballquery_kernel<<<dim3(NSAMP / (BQ_WAVES * 16), BATCH), BQ_WAVES * 32, 0, stream>>>(
      xyz, out, ball_idx);

  // 4) fused gather + max-pool -> tail of d_out
  gather_maxpool_kernel<<<dim3(NSAMP / 32, BATCH), 256, 0, stream>>>(featT, ball_idx, out);
}